// ConvNet01_2929167696049
// MI455X (gfx1250) — compile-verified
//
#include <hip/hip_runtime.h>
#include <hip/hip_bf16.h>

typedef __attribute__((ext_vector_type(2))) float v2f;
typedef __attribute__((ext_vector_type(8))) float v8f;

// Sizes (compile-time constants from the reference)
#define BATCH   4096
#define CIN     3
#define HIN     32
#define WIN     32
#define COUT    16
#define KSZ     5
#define KTOT    75          // CIN*KSZ*KSZ
#define HOUT    28
#define WOUT    28
#define HP      14
#define WP      14
#define FFEAT   3136        // 16*14*14
#define NOUT    10
#define KSTEPS  19          // ceil(76/4); K=76..79 all-zero step dropped

__device__ __forceinline__ float hermite5(float xv,
    float cf0, float cf1, float cf2, float cf3, float cf4, float cf5) {
  // Clenshaw for physicists' Hermite, n = 6 coefficients (matches hermval)
  float x2 = 2.0f * xv;
  float c0 = cf4, c1 = cf5, tmp;
  tmp = c0; c0 = cf3 - c1 * 8.0f; c1 = tmp + c1 * x2;   // nd=5
  tmp = c0; c0 = cf2 - c1 * 6.0f; c1 = tmp + c1 * x2;   // nd=4
  tmp = c0; c0 = cf1 - c1 * 4.0f; c1 = tmp + c1 * x2;   // nd=3
  tmp = c0; c0 = cf0 - c1 * 2.0f; c1 = tmp + c1 * x2;   // nd=2
  return c0 + c1 * x2;
}

// ---------------------------------------------------------------------------
// Kernel 1: im2col-GEMM conv (f32 WMMA) + bias + Hermite + 2x2 maxpool.
// One wave = one 16x16 WMMA tile: M = 4x4 spatial patch, N = 16 channels.
// 49 tiles per image (7x7 patches of 4x4 over 28x28). 8 tiles per wave.
// Grid: 3136 blocks x 256 threads (8 waves) -> 25088 waves x 8 tiles = 200704.
// ---------------------------------------------------------------------------
__global__ __launch_bounds__(256) void conv_herm_pool_kernel(
    const float* __restrict__ x,       // [B,3,32,32]
    const float* __restrict__ coef,    // [6]
    const float* __restrict__ conv_w,  // [16,3,5,5]
    const float* __restrict__ conv_b,  // [16]
    float* __restrict__ pooled)        // [B,16,14,14]
{
  const int lane = threadIdx.x & 31;
  const int wave = threadIdx.x >> 5;
  const int n    = lane & 15;            // N column / A row within tile
  const int khi  = (lane >> 4) * 2;      // K sub-offset per lane half

  // --- Preload B fragments (weights) for all 19 K-steps: B[k][n]=conv_w[n][k]
  // Layout guess (mirrors A 16x4): VGPR0: lanes0-15 K=kb+0, lanes16-31 K=kb+2;
  // VGPR1: K=kb+1 / K=kb+3  -> per-lane pair (kb+khi, kb+khi+1).
  v2f bf[KSTEPS];
  int offA0[KSTEPS], offA1[KSTEPS];
#pragma unroll
  for (int s = 0; s < KSTEPS; ++s) {
    int k0 = 4 * s + khi;
    int k1 = k0 + 1;
    bf[s].x = (k0 < KTOT) ? conv_w[n * KTOT + k0] : 0.0f;
    bf[s].y = (k1 < KTOT) ? conv_w[n * KTOT + k1] : 0.0f;
    // im2col input offsets: k -> (c, kh, kw); clamp OOB k to 0 (B is 0 there)
    int kc0 = (k0 < KTOT) ? k0 : 0;
    int kc1 = (k1 < KTOT) ? k1 : 0;
    int c0 = kc0 / 25, r0 = kc0 - 25 * c0, kh0 = r0 / 5, kw0 = r0 - 5 * kh0;
    int c1 = kc1 / 25, r1 = kc1 - 25 * c1, kh1 = r1 / 5, kw1 = r1 - 5 * kh1;
    offA0[s] = c0 * (HIN * WIN) + kh0 * WIN + kw0;
    offA1[s] = c1 * (HIN * WIN) + kh1 * WIN + kw1;
  }

  const float bias = conv_b[n];
  const float cf0 = coef[0], cf1 = coef[1], cf2 = coef[2];
  const float cf3 = coef[3], cf4 = coef[4], cf5 = coef[5];

  const int gwave = blockIdx.x * 8 + wave;
  const int tileBase = gwave * 8;
  const int m = lane & 15;               // A row: pixel index in 4x4 patch

  for (int ti = 0; ti < 8; ++ti) {
    const int tile = tileBase + ti;
    const int b  = tile / 49;
    const int t  = tile - 49 * b;
    const int row0 = (t / 7) * 4;
    const int col0 = (t - 7 * (t / 7)) * 4;
    const int prow = row0 + (m >> 2);
    const int pcol = col0 + (m & 3);
    const float* xb = x + b * (CIN * HIN * WIN);
    const int pixBase = prow * WIN + pcol;

    v8f acc = {};
#pragma unroll
    for (int s = 0; s < KSTEPS; ++s) {
      v2f a;
      a.x = xb[pixBase + offA0[s]];
      a.y = xb[pixBase + offA1[s]];
      acc = __builtin_amdgcn_wmma_f32_16x16x4_f32(
          false, a, false, bf[s], (short)0, acc, false, false);
    }

    // bias + Hermite activation on all 8 accumulator elements
    float d[8];
#pragma unroll
    for (int v = 0; v < 8; ++v)
      d[v] = hermite5(acc[v] + bias, cf0, cf1, cf2, cf3, cf4, cf5);

    // 2x2 maxpool: per-lane max over VGPRs {0,1,4,5} and {2,3,6,7}.
    // lanes 0-15 -> pooled row row0/2 ; lanes 16-31 -> row0/2 + 1
    float p0 = fmaxf(fmaxf(d[0], d[1]), fmaxf(d[4], d[5]));
    float p1 = fmaxf(fmaxf(d[2], d[3]), fmaxf(d[6], d[7]));
    const int prr = (row0 >> 1) + (lane >> 4);
    const int pcc = (col0 >> 1);
    float* op = pooled + ((b * COUT + n) * HP + prr) * WP + pcc;
    op[0] = p0;
    op[1] = p1;
  }
}

// ---------------------------------------------------------------------------
// Kernel 2: FC [4096,3136] x [3136,10] + bias via f32 WMMA (N padded to 16).
// One block (8 waves) per 16-batch M-tile; 784 K-steps split across waves,
// partials reduced through LDS. Grid: 256 blocks x 256 threads.
// ---------------------------------------------------------------------------
__global__ __launch_bounds__(256) void fc_wmma_kernel(
    const float* __restrict__ pooled,  // [B, 3136]
    const float* __restrict__ fc_w,    // [10, 3136]
    const float* __restrict__ fc_b,    // [10]
    float* __restrict__ out)           // [B, 10]
{
  __shared__ float red[8][8][32];      // [wave][vgpr][lane] = 8 KB
  const int lane = threadIdx.x & 31;
  const int wave = threadIdx.x >> 5;
  const int tile = blockIdx.x;         // M-tile of 16 batches
  const int n    = lane & 15;
  const int khi  = (lane >> 4) * 2;
  const int m    = lane & 15;

  const float* arow = pooled + (size_t)(tile * 16 + m) * FFEAT;
  const bool   bval = (n < NOUT);
  const float* brow = fc_w + n * FFEAT;

  v8f acc = {};
  for (int s = wave; s < FFEAT / 4; s += 8) {
    const int k0 = 4 * s + khi;
    v2f a, bb;
    a.x  = arow[k0];
    a.y  = arow[k0 + 1];
    bb.x = bval ? brow[k0]     : 0.0f;
    bb.y = bval ? brow[k0 + 1] : 0.0f;
    acc = __builtin_amdgcn_wmma_f32_16x16x4_f32(
        false, a, false, bb, (short)0, acc, false, false);
  }

#pragma unroll
  for (int v = 0; v < 8; ++v) red[wave][v][lane] = acc[v];
  __syncthreads();

  if (wave == 0) {
#pragma unroll
    for (int v = 0; v < 8; ++v) {
      float sum = 0.0f;
#pragma unroll
      for (int w = 0; w < 8; ++w) sum += red[w][v][lane];
      if (n < NOUT) {
        const int batch = tile * 16 + v + (lane >> 4) * 8;  // M = v or v+8
        out[batch * NOUT + n] = sum + fc_b[n];
      }
    }
  }
}

extern "C" void kernel_launch(void* const* d_in, const int* in_sizes, int n_in,
                              void* d_out, int out_size, void* d_ws, size_t ws_size,
                              hipStream_t stream) {
  const float* x      = (const float*)d_in[0];
  const float* coef   = (const float*)d_in[1];
  const float* conv_w = (const float*)d_in[2];
  const float* conv_b = (const float*)d_in[3];
  const float* fc_w   = (const float*)d_in[4];
  const float* fc_b   = (const float*)d_in[5];
  float* out    = (float*)d_out;
  float* pooled = (float*)d_ws;   // needs 4096*3136*4 = 51,380,224 bytes

  // 4096 images * 49 tiles = 200704 tiles; 8 tiles/wave, 8 waves/block
  conv_herm_pool_kernel<<<3136, 256, 0, stream>>>(x, coef, conv_w, conv_b, pooled);
  fc_wmma_kernel<<<256, 256, 0, stream>>>(pooled, fc_w, fc_b, out);
}